// KrausOperator_88527865905688
// MI455X (gfx1250) — compile-verified
//
#include <hip/hip_runtime.h>
#include <cmath>

#define DIM        64
#define KSTR       65            // padded LDS row stride (floats) -> conflict-free transposed reads
#define NKRAUS     8
#define POWER_ITERS 200
#define NTHREADS   256           // 8 wave32 -> 16 tiles of 16x16, 2 per wave

typedef __attribute__((ext_vector_type(2))) float v2f;
typedef __attribute__((ext_vector_type(8))) float v8f;

__device__ __forceinline__ v8f wmma4(v2f a, v2f b, v8f c) {
  // V_WMMA_F32_16X16X4_F32 : D = A(16x4) * B(4x16) + C(16x16), all f32
  return __builtin_amdgcn_wmma_f32_16x16x4_f32(
      /*neg_a=*/false, a, /*neg_b=*/false, b,
      /*c_mod=*/(short)0, c, /*reuse_a=*/false, /*reuse_b=*/false);
}

// One wave computes two 16x16 output tiles (tm, tn0) and (tm, tn0+1) of
// D = op(X) * op(Y), 64x64x64, operands read from stride-KSTR LDS buffers.
// XT: A[i][k] = X[k][i]
// YT: B[k][j] = Y[j][k]  (contiguous v2f components -> single ds_load_b64)
template<bool XT, bool YT>
__device__ __forceinline__ void wave_mm(const float* X, const float* Y,
                                        int lane, int tm, int tn0, v8f& c0, v8f& c1) {
  const int r    = lane & 15;
  const int koff = (lane >> 4) << 1;          // lanes 16..31 hold K+2 (A) / rows K+2 (B)
  const int rowA = tm * 16 + r;
  const int col0 = tn0 * 16 + r;
  const int col1 = col0 + 16;
#pragma unroll
  for (int kk = 0; kk < DIM / 4; ++kk) {
    const int k0 = kk * 4 + koff;
    v2f a, b0, b1;
    if (XT) { a.x = X[(k0    ) * KSTR + rowA];
              a.y = X[(k0 + 1) * KSTR + rowA]; }
    else    { a.x = X[rowA * KSTR + (k0    )];
              a.y = X[rowA * KSTR + (k0 + 1)]; }
    if (YT) { b0.x = Y[col0 * KSTR + (k0    )]; b0.y = Y[col0 * KSTR + (k0 + 1)];
              b1.x = Y[col1 * KSTR + (k0    )]; b1.y = Y[col1 * KSTR + (k0 + 1)]; }
    else    { b0.x = Y[(k0    ) * KSTR + col0]; b0.y = Y[(k0 + 1) * KSTR + col0];
              b1.x = Y[(k0    ) * KSTR + col1]; b1.y = Y[(k0 + 1) * KSTR + col1]; }
    c0 = wmma4(a, b0, c0);     // independent accumulators -> co-issueable WMMAs
    c1 = wmma4(a, b1, c1);
  }
}

// Store a 16x16 f32 C/D tile (8 VGPRs per lane) into a stride-KSTR LDS buffer,
// optionally scaled (folds the pending trace normalization, 8 muls per tile).
__device__ __forceinline__ void store_tile(float* D, int lane, int tm, int tn,
                                           const v8f& c, float scale) {
  const int col   = tn * 16 + (lane & 15);
  const int rbase = tm * 16 + ((lane >> 4) << 3);   // upper half-lanes hold M+8
#pragma unroll
  for (int i = 0; i < 8; ++i) D[(rbase + i) * KSTR + col] = c[i] * scale;
}

// rhoOut (unscaled) = sum_{a in [aFirst, aFirst+aCount)} op(K_a) * (rho*binv) * op(K_a)
// LEFT=false:  K_a  rho K_a^T      LEFT=true:  K_a^T rho K_a
// rho is symmetric (density matrix), so its B operand is read with the
// TRANSPOSED pattern (YT=true): contiguous pairs, zero repacking moves.
// binv is folded into the T-tile stores: (Ka*rho)*binv == Ka*(rho*binv).
// Writes per-tile-row partial traces of the (unscaled) result into pxPart[0..3].
template<bool LEFT>
__device__ __forceinline__ void cp_apply(const float* Kl, float* rho, float* T,
                                         float* rhoOut, float* pxPart, float binv,
                                         int aFirst, int aCount, int lane, int wave) {
  const int tm  = wave >> 1;
  const int tn0 = (wave & 1) * 2;
  v8f c0 = {}; v8f c1 = {};
#pragma unroll 1
  for (int ai = 0; ai < aCount; ++ai) {
    const float* Ka = Kl + (size_t)(aFirst + ai) * (DIM * KSTR);
    {   // mm1:  T = (op(Ka) * rho) * binv     (rho read transposed == rho, symmetric)
      v8f t0 = {}; v8f t1 = {};
      wave_mm<LEFT, true>(Ka, rho, lane, tm, tn0, t0, t1);
      store_tile(T, lane, tm, tn0,     t0, binv);
      store_tile(T, lane, tm, tn0 + 1, t1, binv);
    }
    __syncthreads();
    // mm2:  c += T * op2(Ka)
    wave_mm<false, !LEFT>(T, Ka, lane, tm, tn0, c0, c1);
    if (ai != aCount - 1) __syncthreads();   // protect T before next overwrite
  }
  // Partial trace from diagonal tiles (waves 0,2,5,7 own them).
  if (tm == tn0 || tm == tn0 + 1) {
    v8f cd = (tm == tn0) ? c0 : c1;
    float d = 0.0f;
#pragma unroll
    for (int i = 0; i < 8; ++i) {           // lane i (i<8) holds C[i][i] in vgpr i,
      float vi = cd[i];                     // lane i+24 holds C[i+8][i+8] in vgpr i
      d += ((lane == i) || (lane == i + 24)) ? vi : 0.0f;
    }
#pragma unroll
    for (int off = 16; off > 0; off >>= 1) d += __shfl_xor(d, off, 32);
    if (lane == 0) pxPart[tm] = d;
  }
  store_tile(rhoOut, lane, tm, tn0,     c0, 1.0f);   // unscaled; scale pending
  store_tile(rhoOut, lane, tm, tn0 + 1, c1, 1.0f);
}

// trace(A @ B) over stride-KSTR LDS matrices; red needs >= 257 floats (T buffer reused).
__device__ __forceinline__ float block_tr_dot(const float* A, const float* B,
                                              float* red, int tid) {
  float s = 0.0f;
  for (int idx = tid; idx < DIM * DIM; idx += NTHREADS) {
    const int i = idx >> 6, k = idx & 63;
    s += A[i * KSTR + k] * B[k * KSTR + i];
  }
  red[tid] = s;
  __syncthreads();
  if (tid == 0) { float t = 0.0f; for (int j = 0; j < NTHREADS; ++j) t += red[j]; red[NTHREADS] = t; }
  __syncthreads();
  float r = red[NTHREADS];
  __syncthreads();
  return r;
}

// Block 0: right fixed point.  Block 1: left fixed point (+ lambda).
// ws[0..4095] = rhor, ws[4096..8191] = rhol, ws[8192] = lambda (all normalized).
__global__ void __launch_bounds__(NTHREADS)
kraus_power_kernel(const float* __restrict__ K, float* __restrict__ ws) {
  extern __shared__ char smem[];
  float* Kl   = (float*)smem;                              // 8*64*65
  float* T    = Kl  + NKRAUS * DIM * KSTR;                 // 64*65
  float* RHO  = T   + DIM * KSTR;                          // 64*65
  float* scal = RHO + DIM * KSTR;                          // [0..3] = pxPart
  const int tid = threadIdx.x, lane = tid & 31, wave = tid >> 5;

  for (int i = tid; i < NKRAUS * DIM * DIM; i += NTHREADS) {
    const int a = i >> 12, rc = i & 4095, r = rc >> 6, c = rc & 63;
    Kl[(size_t)a * DIM * KSTR + r * KSTR + c] = K[i];
  }
  for (int i = tid; i < DIM * DIM; i += NTHREADS) {
    const int r = i >> 6, c = i & 63;
    RHO[r * KSTR + c] = (r == c) ? (1.0f / DIM) : 0.0f;    // eye/dim
  }
  __syncthreads();

  const bool left = (blockIdx.x == 1);
  float binv = 1.0f, mu = 1.0f;                            // per-thread (uniform) scalars
#pragma unroll 1
  for (int it = 0; it < POWER_ITERS; ++it) {
    if (left) cp_apply<true >(Kl, RHO, T, RHO, scal, binv, 0, NKRAUS, lane, wave);
    else      cp_apply<false>(Kl, RHO, T, RHO, scal, binv, 0, NKRAUS, lane, wave);
    __syncthreads();                                       // pxPart visible to all
    mu   = (scal[0] + scal[1]) + (scal[2] + scal[3]);      // all threads, no 2nd barrier
    binv = 1.0f / mu;
  }
  float* out = ws + (size_t)blockIdx.x * (DIM * DIM);
  for (int i = tid; i < DIM * DIM; i += NTHREADS) {
    const int r = i >> 6, c = i & 63;
    out[i] = RHO[r * KSTR + c] * binv;                     // apply pending scale
  }
  if (left && tid == 0) ws[2 * DIM * DIM] = mu;            // lambda = last left mu
}

__global__ void __launch_bounds__(NTHREADS)
kraus_scan_kernel(const float* __restrict__ K, const int* __restrict__ seq,
                  const float* __restrict__ ws, float* __restrict__ out, int L) {
  extern __shared__ char smem[];
  float* Kl   = (float*)smem;                              // normalized Kraus (stride 65)
  float* T    = Kl   + NKRAUS * DIM * KSTR;
  float* RHO  = T    + DIM * KSTR;
  float* RHOL = RHO  + DIM * KSTR;
  float* scal = RHOL + DIM * KSTR;                         // [0..3] = pxPart
  int*   SEQ  = (int*)(scal + 32);
  const int tid = threadIdx.x, lane = tid & 31, wave = tid >> 5;

  const float kscale = rsqrtf(ws[2 * DIM * DIM]);          // Kn = K / sqrt(lambda)
  for (int i = tid; i < NKRAUS * DIM * DIM; i += NTHREADS) {
    const int a = i >> 12, rc = i & 4095, r = rc >> 6, c = rc & 63;
    Kl[(size_t)a * DIM * KSTR + r * KSTR + c] = K[i] * kscale;
  }
  for (int i = tid; i < DIM * DIM; i += NTHREADS) {
    const int r = i >> 6, c = i & 63;
    RHO [r * KSTR + c] = ws[i];                            // rhor
    RHOL[r * KSTR + c] = ws[DIM * DIM + i];                // rhol
  }
  for (int i = tid; i < L; i += NTHREADS) SEQ[i] = seq[i];
  __syncthreads();

  const float tr0 = block_tr_dot(RHO, RHOL, T, tid);       // trace(rhor @ rhol)
  float binv = 1.0f / tr0;                                 // rho0 = rhor/tr0 (pending)
  float logp = 0.0f;                                       // per-thread (uniform) log2-prob

#pragma unroll 1
  for (int t = 0; t < L; ++t) {
    const int a = SEQ[t];                                  // uniform LDS broadcast
    cp_apply<false>(Kl, RHO, T, RHO, scal, binv, a, 1, lane, wave);
    __syncthreads();                                       // only loop barrier: 2/step total
    const float px = (scal[0] + scal[1]) + (scal[2] + scal[3]);
    binv  = 1.0f / px;                                     // redundant in all threads:
    logp -= log2f(px);                                     // kills scalar-section barriers
  }

  // Final correction: logp -= log2(trace(rho_f_norm @ rhol)), rho_f still unscaled.
  const float dot = block_tr_dot(RHO, RHOL, T, tid);
  if (tid == 0) out[0] = logp - log2f(dot * binv);
}

extern "C" void kernel_launch(void* const* d_in, const int* in_sizes, int n_in,
                              void* d_out, int out_size, void* d_ws, size_t ws_size,
                              hipStream_t stream) {
  const float* K   = (const float*)d_in[0];   // (8, 64, 64) f32
  const int*   seq = (const int*)d_in[1];     // (L,) i32
  const int    L   = in_sizes[1];
  float* out = (float*)d_out;
  float* ws  = (float*)d_ws;                  // 2*4096 + 1 floats used

  const size_t smemP = (size_t)(NKRAUS * DIM * KSTR + 2 * DIM * KSTR + 32) * sizeof(float);
  const size_t smemS = (size_t)(NKRAUS * DIM * KSTR + 3 * DIM * KSTR + 32) * sizeof(float)
                     + (size_t)L * sizeof(int);

  kraus_power_kernel<<<dim3(2), dim3(NTHREADS), smemP, stream>>>(K, ws);
  kraus_scan_kernel <<<dim3(1), dim3(NTHREADS), smemS, stream>>>(K, seq, ws, out, L);
}